// SNNLayer_43035572306630
// MI455X (gfx1250) — compile-verified
//
#include <hip/hip_runtime.h>

// SNN layer: spikes = clip(pre_spikes @ W + const_inp - thr, 0, 0.9)
// B=8 batches (padded to 16 in WMMA M dim), M=N=8192.
//
// Roofline: W = 256 MB streamed exactly once -> ~11.5 us floor at 23.3 TB/s;
// only ~93 TFLOPS fp32 needed, so V_WMMA_F32_16X16X4_F32 is far from its
// ceiling. Strategy: widest possible non-temporal W loads (b128, NT so the
// 256 MB stream doesn't thrash the 192 MB L2), 4 WMMAs per 2 W-loads, and
// enough workgroups (512) that no WGP is idle for the bandwidth-bound stream.

typedef float v2f __attribute__((ext_vector_type(2)));
typedef float v4f __attribute__((ext_vector_type(4)));
typedef float v8f __attribute__((ext_vector_type(8)));

#define BATCH   8
#define MDIM    8192
#define NDIM    8192
#define WAVES   16                 // waves per block (in-block split-K ways)
#define THREADS (WAVES * 32)       // 512
#define TILES   4                  // 16-col WMMA tiles per wave
#define COLS    (16 * TILES)       // 64 columns per block
#define KSPLIT  4                  // K-slices across blocks (grid.y)
#define SPIKE_CAP 0.9f

#define WMMA_F32(A, Bf, C) \
    __builtin_amdgcn_wmma_f32_16x16x4_f32(false, (A), false, (Bf), (short)0, (C), false, false)

// PARTIAL=true : write 8x64 partial tile to ws[kslice][8][NDIM] (phase 1 of 2)
// PARTIAL=false: single pass, fused epilogue, write out directly
template <int KS, bool PARTIAL>
__global__ __launch_bounds__(THREADS) void snn_wmma_kernel(
    const float* __restrict__ pre,   // [8, 8192]
    const float* __restrict__ W,     // [8192, 8192] row-major
    const float* __restrict__ thr,   // [8, 8192]
    const float* __restrict__ cinp,  // [8, 8192]
    float* __restrict__ out,         // [8, 8192]
    float* __restrict__ ws)          // [KS, 8, 8192] partials (PARTIAL only)
{
    __shared__ float red[TILES][WAVES][256];   // 64 KB of partial tiles

    constexpr int KPW = MDIM / KS / WAVES;     // K per wave (compile-time)

    const int tid  = threadIdx.x;
    const int wave = tid >> 5;
    const int lane = tid & 31;
    const int n0   = blockIdx.x * COLS;
    const int kb   = blockIdx.y * (MDIM / KS); // this block's K slice

    // WMMA lane roles (wave32):
    //  A 16x4 f32: lanes 0-15 hold K={0,1}; lanes 16-31 hold K={2,3}
    //  B 4x16 f32: VGPR0 = row (k+q), VGPR1 = row (k+q+1); N index = lane%16
    const int m16  = lane & 15;          // M slot (batch; 8..15 zero-pad)
    const int q    = (lane >> 4) * 2;    // K sub-offset this lane supplies

    const bool avalid = (m16 < BATCH);
    const float* aptr = pre + (size_t)(avalid ? m16 : 0) * MDIM;
    // Each lane streams 4 consecutive W columns: tile t owns column n0+4*m16+t.
    const float* wptr = W + (size_t)q * NDIM + n0 + 4 * m16;

    const int k0 = kb + wave * KPW;      // this wave's split-K range
    v8f acc0 = {}, acc1 = {}, acc2 = {}, acc3 = {};

    #pragma unroll 4
    for (int i = 0; i < KPW / 4; ++i) {  // compile-time trip count -> scalar loop
        const int k = k0 + i * 4;

        // A fragment: pre[m16][k+q], pre[m16][k+q+1]  (aligned b64, L2-resident)
        v2f a = *(const v2f*)(aptr + k + q);
        a.x = avalid ? a.x : 0.0f;       // zero pad rows; no EXEC divergence
        a.y = avalid ? a.y : 0.0f;

        // W: two non-temporal b128 loads cover rows k+q and k+q+1,
        // 4 consecutive columns each (256 contiguous bytes per half-wave).
        const v4f w0 = __builtin_nontemporal_load((const v4f*)(wptr + (size_t)k * NDIM));
        const v4f w1 = __builtin_nontemporal_load((const v4f*)(wptr + (size_t)k * NDIM + NDIM));

        const v2f b0 = { w0.x, w1.x };   // tile 0: columns n0 + 4m + 0
        const v2f b1 = { w0.y, w1.y };   // tile 1
        const v2f b2 = { w0.z, w1.z };   // tile 2
        const v2f b3 = { w0.w, w1.w };   // tile 3

        acc0 = WMMA_F32(a, b0, acc0);
        acc1 = WMMA_F32(a, b1, acc1);
        acc2 = WMMA_F32(a, b2, acc2);
        acc3 = WMMA_F32(a, b3, acc3);
    }

    // Dump this wave's four 16x16 partial tiles to LDS.
    #pragma unroll
    for (int r = 0; r < 8; ++r) {
        red[0][wave][r * 32 + lane] = acc0[r];
        red[1][wave][r * 32 + lane] = acc1[r];
        red[2][wave][r * 32 + lane] = acc2[r];
        red[3][wave][r * 32 + lane] = acc3[r];
    }

    __syncthreads();

    // In-block split-K reduction: 1024 tile elements over 512 threads.
    #pragma unroll
    for (int e = tid; e < TILES * 256; e += THREADS) {
        const int t   = e >> 8;          // tile
        const int idx = e & 255;         // element within 16x16 tile

        float s = 0.0f;
        #pragma unroll
        for (int w = 0; w < WAVES; ++w)
            s += red[t][w][idx];

        // C/D layout: VGPR r, lanes 0-15 -> M=r; lanes 16-31 -> M=r+8; N=lane%16
        const int r = idx >> 5;
        const int l = idx & 31;
        const int m = r + ((l >> 4) << 3);
        const int c = n0 + ((l & 15) << 2) + t;
        if (m < BATCH) {
            const size_t o = (size_t)m * NDIM + c;
            if (PARTIAL) {
                ws[(size_t)blockIdx.y * BATCH * NDIM + o] = s;
            } else {
                float x = s + cinp[o] - thr[o];
                out[o] = fminf(fmaxf(x, 0.0f), SPIKE_CAP);
            }
        }
    }
}

// Phase 2: sum KSPLIT partials and apply the spike epilogue.
__global__ __launch_bounds__(256) void snn_reduce_kernel(
    const float* __restrict__ ws,    // [KSPLIT, 8, 8192]
    const float* __restrict__ thr,
    const float* __restrict__ cinp,
    float* __restrict__ out)         // [8, 8192]
{
    const size_t o = (size_t)blockIdx.x * 256 + threadIdx.x;  // < BATCH*NDIM
    float s = 0.0f;
    #pragma unroll
    for (int y = 0; y < KSPLIT; ++y)
        s += ws[(size_t)y * BATCH * NDIM + o];
    float x = s + cinp[o] - thr[o];
    out[o] = fminf(fmaxf(x, 0.0f), SPIKE_CAP);
}

extern "C" void kernel_launch(void* const* d_in, const int* in_sizes, int n_in,
                              void* d_out, int out_size, void* d_ws, size_t ws_size,
                              hipStream_t stream) {
    const float* pre  = (const float*)d_in[0];   // pre_spikes [8,1,8192]
    const float* W    = (const float*)d_in[1];   // W [8192,8192]
    const float* thr  = (const float*)d_in[2];   // thr [8,1,8192]
    const float* cinp = (const float*)d_in[3];   // const_inp [8,1,8192]
    float* out = (float*)d_out;                  // spikes [8,1,8192]

    const size_t ws_needed = (size_t)KSPLIT * BATCH * NDIM * sizeof(float);  // 1 MB

    if (ws_size >= ws_needed) {
        // 512 workgroups: 128 column tiles x 4 K-slices -> fills all WGPs.
        dim3 grid(NDIM / COLS, KSPLIT);
        snn_wmma_kernel<KSPLIT, true><<<grid, dim3(THREADS), 0, stream>>>(
            pre, W, thr, cinp, out, (float*)d_ws);
        snn_reduce_kernel<<<dim3((BATCH * NDIM) / 256), dim3(256), 0, stream>>>(
            (const float*)d_ws, thr, cinp, out);
    } else {
        // Fallback: single pass, fused epilogue (128 workgroups).
        dim3 grid(NDIM / COLS, 1);
        snn_wmma_kernel<1, false><<<grid, dim3(THREADS), 0, stream>>>(
            pre, W, thr, cinp, out, nullptr);
    }
}